// ModernAttentionPlatonic_47158740910518
// MI455X (gfx1250) — compile-verified
//
#include <hip/hip_runtime.h>
#include <math.h>

typedef float v2f __attribute__((ext_vector_type(2)));
typedef float v8f __attribute__((ext_vector_type(8)));

#define G_   12
#define H_   8
#define BN_  1024          // B*N = 2*512
#define CH_  3072          // G*H*32

// D = A(16x4) * B(4x16) + C, full f32 precision (CDNA5 V_WMMA_F32_16X16X4_F32)
static __device__ __forceinline__ v8f wmma_f32_4(v2f a, v2f b, v8f c) {
  return __builtin_amdgcn_wmma_f32_16x16x4_f32(false, a, false, b, (short)0, c,
                                               false, false);
}

// ---------------------------------------------------------------------------
// Generic platonic (block-circulant) projection:
//   out[bn, h*co + o] = sum_g sum_i x[bn, g*ci + i] * W[(g-h)%G][o][i]
// grid = (G, co/64, BN/64), block = 128 (4 waves). 64x64 output tile/block.
// ---------------------------------------------------------------------------
__global__ __launch_bounds__(128)
void platonic_proj_kernel(const float* __restrict__ x, const float* __restrict__ W,
                          float* __restrict__ out, int ci, int co) {
  __shared__ float xs[64][65];   // padded: conflict-free fragment reads
  __shared__ float wsm[64][65];
  const int tid  = threadIdx.x;
  const int wave = tid >> 5;
  const int lane = tid & 31;
  const int half = lane >> 4;
  const int llo  = lane & 15;
  const int h    = blockIdx.x;
  const int ob   = blockIdx.y * 64;
  const int bnb  = blockIdx.z * 64;
  const int xstride = G_ * ci;

  v8f acc[4];
  #pragma unroll
  for (int t = 0; t < 4; ++t)
    #pragma unroll
    for (int j = 0; j < 8; ++j) acc[t][j] = 0.0f;

  for (int g = 0; g < G_; ++g) {
    const int wg = (g - h + G_) % G_;
    const float* Wg = W + (size_t)wg * co * ci;
    for (int ic = 0; ic < ci; ic += 64) {
      #pragma unroll 4
      for (int it = 0; it < 32; ++it) {          // 4096 elements, coalesced rows
        const int idx = it * 128 + tid;
        const int r = idx >> 6, c = idx & 63;
        xs[r][c]  = x[(size_t)(bnb + r) * xstride + g * ci + ic + c];
        wsm[r][c] = Wg[(size_t)(ob + r) * ci + ic + c];
      }
      __syncthreads();
      #pragma unroll
      for (int kk = 0; kk < 64; kk += 4) {
        v2f a;                                   // A[m][k], m = lane%16
        a.x = xs[wave * 16 + llo][kk + 2 * half + 0];
        a.y = xs[wave * 16 + llo][kk + 2 * half + 1];
        #pragma unroll
        for (int t = 0; t < 4; ++t) {
          v2f b;                                 // B[k][n] = W[o=n][i=k]
          b.x = wsm[t * 16 + llo][kk + 2 * half + 0];
          b.y = wsm[t * 16 + llo][kk + 2 * half + 1];
          acc[t] = wmma_f32_4(a, b, acc[t]);
        }
      }
      __syncthreads();
    }
  }
  const int ostride = G_ * co;
  #pragma unroll
  for (int t = 0; t < 4; ++t)
    #pragma unroll
    for (int r = 0; r < 8; ++r)
      out[(size_t)(bnb + wave * 16 + 8 * half + r) * ostride +
          h * co + ob + t * 16 + llo] = acc[t][r];
}

// ---------------------------------------------------------------------------
// Fused rms_norm + sequence RoPE + platonic RoPE (rotations compose: one angle)
// One thread per pair f in one head; 16 lanes (half-wave) per head.
// qk layout: [BN][96 heads][32]
// ---------------------------------------------------------------------------
__global__ __launch_bounds__(256)
void norm_rope_kernel(float* __restrict__ qk, const float* __restrict__ nw,
                      const float* __restrict__ coords,
                      const float* __restrict__ freqs,
                      const int* __restrict__ seq) {
  const int tid  = blockIdx.x * 256 + threadIdx.x;
  const int f    = tid & 15;                 // pair index 0..15
  const int head = tid >> 4;                 // bn*96 + gh
  const int bn   = head / 96;
  const int gh   = head % 96;
  const int g    = gh >> 3;
  const int hh   = gh & 7;

  float2* p2 = reinterpret_cast<float2*>(qk) + (size_t)head * 16;
  const float2 xv = p2[f];

  // rms_norm: sum of squares across the 16 lanes of this half-wave
  float ss = xv.x * xv.x + xv.y * xv.y;
  #pragma unroll
  for (int m = 1; m < 16; m <<= 1) ss += __shfl_xor(ss, m, 32);
  const float rinv = rsqrtf(ss * (1.0f / 32.0f) + 1.1920928955078125e-7f);
  const float2 wv = reinterpret_cast<const float2*>(nw)[f];
  const float x1 = xv.x * rinv * wv.x;
  const float x2 = xv.y * rinv * wv.y;

  // sequence rope angle
  const float pos  = (float)seq[bn];
  const float invf = __powf(10000.0f, -(float)f * (1.0f / 16.0f));

  // platonic rope angle: phase = (coords . Rz(2*pi*g/G)) . freqs[hh,f,:]
  float sg, cg;
  __sincosf(6.28318530717958648f * (float)g * (1.0f / 12.0f), &sg, &cg);
  const float cx = coords[bn * 3 + 0];
  const float cy = coords[bn * 3 + 1];
  const float cz = coords[bn * 3 + 2];
  const float u0 =  cx * cg + cy * sg;
  const float u1 = -cx * sg + cy * cg;
  const float* F = freqs + (size_t)(hh * 16 + f) * 3;
  const float phase = u0 * F[0] + u1 * F[1] + cz * F[2];

  float sa, ca;
  __sincosf(pos * invf + phase, &sa, &ca);
  float2 ov;
  ov.x = x1 * ca - x2 * sa;
  ov.y = x1 * sa + x2 * ca;
  p2[f] = ov;
}

// ---------------------------------------------------------------------------
// Flash attention per head. grid = (192 heads, 8 q-tiles of 64), block = 128.
// Each wave owns 16 q rows; keys streamed through LDS in chunks of 64.
// ---------------------------------------------------------------------------
__global__ __launch_bounds__(128)
void attn_kernel(const float* __restrict__ Q, const float* __restrict__ K,
                 const float* __restrict__ V, float* __restrict__ O) {
  __shared__ float ks[64][33];
  __shared__ float vs[64][33];
  __shared__ float ps[4][16][17];              // per-wave P transpose scratch
  const int tid  = threadIdx.x;
  const int wave = tid >> 5;
  const int lane = tid & 31;
  const int half = lane >> 4;
  const int llo  = lane & 15;
  const int head = blockIdx.x;                 // 0..191
  const int b    = head / 96;
  const int gh   = head % 96;
  const int rowbase = b * 512;
  const int colbase = gh * 32;
  const int qr   = blockIdx.y * 64 + wave * 16;

  // preload this wave's Q fragments (16 rows x 32 k)
  v2f aq[8];
  #pragma unroll
  for (int kki = 0; kki < 8; ++kki) {
    const float* qp = Q + (size_t)(rowbase + qr + llo) * CH_ + colbase +
                      kki * 4 + 2 * half;
    aq[kki].x = qp[0];
    aq[kki].y = qp[1];
  }

  float m8[8], l8[8];
  v8f o0, o1;
  #pragma unroll
  for (int r = 0; r < 8; ++r) { m8[r] = -1e30f; l8[r] = 0.0f; o0[r] = 0.0f; o1[r] = 0.0f; }
  const float scale = 0.17677669529663687f;    // 1/sqrt(32)

  for (int kc = 0; kc < 512; kc += 64) {
    #pragma unroll 4
    for (int it = 0; it < 16; ++it) {          // stage K,V chunk (coalesced)
      const int idx = it * 128 + tid;
      const int r = idx >> 5, c = idx & 31;
      ks[r][c] = K[(size_t)(rowbase + kc + r) * CH_ + colbase + c];
      vs[r][c] = V[(size_t)(rowbase + kc + r) * CH_ + colbase + c];
    }
    __syncthreads();
    for (int sub = 0; sub < 4; ++sub) {
      const int kb = sub * 16;
      // S = Q * K^T  (16x16, K=32 -> 8 wmma)
      v8f s;
      #pragma unroll
      for (int r = 0; r < 8; ++r) s[r] = 0.0f;
      #pragma unroll
      for (int kki = 0; kki < 8; ++kki) {
        v2f bf;
        bf.x = ks[kb + llo][kki * 4 + 2 * half + 0];
        bf.y = ks[kb + llo][kki * 4 + 2 * half + 1];
        s = wmma_f32_4(aq[kki], bf, s);
      }
      // online softmax update per row
      #pragma unroll
      for (int r = 0; r < 8; ++r) {
        const float sv = s[r] * scale;
        float tm = sv;
        #pragma unroll
        for (int m = 1; m < 16; m <<= 1) tm = fmaxf(tm, __shfl_xor(tm, m, 32));
        const float nm = fmaxf(m8[r], tm);
        const float sc = __expf(m8[r] - nm);
        const float p  = __expf(sv - nm);
        float rs = p;
        #pragma unroll
        for (int m = 1; m < 16; m <<= 1) rs += __shfl_xor(rs, m, 32);
        l8[r] = l8[r] * sc + rs;
        m8[r] = nm;
        o0[r] *= sc;
        o1[r] *= sc;
        ps[wave][8 * half + r][llo] = p;       // same-wave LDS transpose
      }
      // O += P(16x16) * V(16x32)  (2 col-tiles, K=16 -> 4 wmma each)
      #pragma unroll
      for (int kki = 0; kki < 4; ++kki) {
        v2f a2;
        a2.x = ps[wave][llo][kki * 4 + 2 * half + 0];
        a2.y = ps[wave][llo][kki * 4 + 2 * half + 1];
        v2f b0, b1;
        b0.x = vs[kb + kki * 4 + 2 * half + 0][llo];
        b0.y = vs[kb + kki * 4 + 2 * half + 1][llo];
        b1.x = vs[kb + kki * 4 + 2 * half + 0][16 + llo];
        b1.y = vs[kb + kki * 4 + 2 * half + 1][16 + llo];
        o0 = wmma_f32_4(a2, b0, o0);
        o1 = wmma_f32_4(a2, b1, o1);
      }
    }
    __syncthreads();
  }
  #pragma unroll
  for (int r = 0; r < 8; ++r) {
    const float inv = 1.0f / l8[r];
    const size_t row = (size_t)(rowbase + qr + 8 * half + r);
    O[row * CH_ + colbase + llo]      = o0[r] * inv;
    O[row * CH_ + colbase + 16 + llo] = o1[r] * inv;
  }
}

// ---------------------------------------------------------------------------
extern "C" void kernel_launch(void* const* d_in, const int* in_sizes, int n_in,
                              void* d_out, int out_size, void* d_ws, size_t ws_size,
                              hipStream_t stream) {
  const float* feat   = (const float*)d_in[0];
  const float* coords = (const float*)d_in[1];
  const float* Wq     = (const float*)d_in[2];
  const float* Wk     = (const float*)d_in[3];
  const float* Wv     = (const float*)d_in[4];
  const float* Wo     = (const float*)d_in[5];
  const float* qnw    = (const float*)d_in[6];
  const float* knw    = (const float*)d_in[7];
  const float* pfreq  = (const float*)d_in[8];
  const int*   seq    = (const int*)d_in[9];

  float* Qb = (float*)d_ws;                 // 1024*3072 f32 each
  float* Kb = Qb + (size_t)BN_ * CH_;
  float* Vb = Kb + (size_t)BN_ * CH_;
  float* Ob = Vb + (size_t)BN_ * CH_;

  const dim3 blk(128);
  // Q/K/V projections (block-circulant GEMM, ci=64, co=256)
  platonic_proj_kernel<<<dim3(12, 4, 16), blk, 0, stream>>>(feat, Wq, Qb, 64, 256);
  platonic_proj_kernel<<<dim3(12, 4, 16), blk, 0, stream>>>(feat, Wk, Kb, 64, 256);
  platonic_proj_kernel<<<dim3(12, 4, 16), blk, 0, stream>>>(feat, Wv, Vb, 64, 256);
  // fused rms_norm + combined rope on Q and K (1024*96 heads * 16 pairs)
  norm_rope_kernel<<<dim3(6144), dim3(256), 0, stream>>>(Qb, qnw, coords, pfreq, seq);
  norm_rope_kernel<<<dim3(6144), dim3(256), 0, stream>>>(Kb, knw, coords, pfreq, seq);
  // flash attention, 192 heads x 8 q-tiles
  attn_kernel<<<dim3(192, 8), blk, 0, stream>>>(Qb, Kb, Vb, Ob);
  // output projection (ci=256, co=64) straight into d_out
  platonic_proj_kernel<<<dim3(12, 1, 16), blk, 0, stream>>>(Ob, Wo, (float*)d_out, 256, 64);
}